// FactorizedEmbedding_12025908429387
// MI455X (gfx1250) — compile-verified
//
#include <hip/hip_runtime.h>

// BlockTT factorized embedding, MI455X (gfx1250, wave32).
// vocab 128000 = 40*50*64, dim 1024 = 8*8*16, rank 8.
//
// Roofline: output stream = 1 GiB (262144 tokens * 4KB) -> ~46us @ 23.3 TB/s.
// Compute = ~6.4 GFLOP fp32 -> trivial. Kernel is store-bound; use f32 WMMA
// (exact precision) for the dominant 64x8 @ 8x16 contraction and NT stores.

typedef __attribute__((ext_vector_type(2))) float v2f;
typedef __attribute__((ext_vector_type(8))) float v8f;

#define TT_M1 50
#define TT_M2 64

__global__ __launch_bounds__(256) void
FactorizedEmbedding_tt_kernel(const int* __restrict__ indices,
                              const float* __restrict__ core0,   // [1,40,8,8]
                              const float* __restrict__ core1,   // [8,50,8,8]
                              const float* __restrict__ core2,   // [8,64,16,1]
                              float* __restrict__ out,           // [T,1024]
                              int T)
{
    // Per-wave private LDS regions (8 waves/block): 18 KB total.
    __shared__ float lds_a1[8][64];        // core0 slice  [n0][r1]
    __shared__ float lds_cur2[8][64][8];   // cur2        [(n0*8+n1)][r2]

    const int wave = threadIdx.x >> 5;
    const int lane = threadIdx.x & 31;
    const int t    = blockIdx.x * 8 + wave;
    if (t >= T) return;                    // wave-uniform: EXEC stays all-1s

    const int idx = indices[t];
    const int i0  = idx / (TT_M1 * TT_M2);            // /3200
    const int rem = idx - i0 * (TT_M1 * TT_M2);
    const int i1  = rem >> 6;                          // /64
    const int i2  = rem & 63;                          // %64

    // ---- stage A1 = core0[0, i0] (8x8 = 64 f32) into LDS ----
    {
        const float* a1g = core0 + (size_t)i0 * 64;
        lds_a1[wave][lane]      = a1g[lane];
        lds_a1[wave][lane + 32] = a1g[lane + 32];
    }
    asm volatile("s_wait_dscnt 0" ::: "memory");       // in-wave LDS fence

    // ---- GEMM1 (VALU): cur2[(n0,n1)][r2] = sum_r1 A1[n0][r1]*core1[r1,i1,n1,r2]
    // Each lane owns 2 of the 64 (n1,r2) columns: 128 FMA/lane.
    {
        const float* c1b = core1 + (size_t)i1 * 64;    // [r1 stride = 50*64]
        #pragma unroll
        for (int pi = 0; pi < 2; ++pi) {
            const int p  = 2 * lane + pi;              // 0..63
            const int n1 = p >> 3;
            const int r2 = p & 7;
            float b1[8];
            #pragma unroll
            for (int r1 = 0; r1 < 8; ++r1)
                b1[r1] = c1b[r1 * (TT_M1 * 64) + n1 * 8 + r2];  // L2-resident
            #pragma unroll
            for (int n0 = 0; n0 < 8; ++n0) {
                float acc = 0.0f;
                #pragma unroll
                for (int r1 = 0; r1 < 8; ++r1)
                    acc = fmaf(lds_a1[wave][n0 * 8 + r1], b1[r1], acc);
                lds_cur2[wave][n0 * 8 + n1][r2] = acc;
            }
        }
    }
    asm volatile("s_wait_dscnt 0" ::: "memory");       // in-wave LDS fence

    // ---- GEMM2 (WMMA f32): D[64x16] = cur2[64x8] @ core2[:, i2][8x16]
    // 4 M-tiles x 2 K-steps of v_wmma_f32_16x16x4_f32 -> exact tiling, no waste.
    const int half = lane >> 4;                        // 0: K={0,1}, 1: K={2,3}
    const int l15  = lane & 15;
    const float* c2b = core2 + (size_t)i2 * 16;        // element [r2,i2,n2]: r2*1024 + n2
    float* outt = out + (size_t)t * 1024;

    #pragma unroll
    for (int m = 0; m < 4; ++m) {
        v8f acc = {};
        #pragma unroll
        for (int k = 0; k < 2; ++k) {
            const int kb = 4 * k + 2 * half;           // K index of VGPR0 slot
            v2f a, b;
            // A 16x4 layout: lanes 0-15 hold K=kb,kb+1 ; lanes 16-31 K=kb(+2),..
            a.x = lds_cur2[wave][16 * m + l15][kb + 0];
            a.y = lds_cur2[wave][16 * m + l15][kb + 1];
            // B 4x16 layout: row K striped over VGPRs, col N = lane&15
            b.x = c2b[(size_t)(kb + 0) * (TT_M2 * 16) + l15];
            b.y = c2b[(size_t)(kb + 1) * (TT_M2 * 16) + l15];
            acc = __builtin_amdgcn_wmma_f32_16x16x4_f32(
                      false, a, false, b, (short)0, acc, false, false);
        }
        // D 16x16 f32 layout: VGPR v -> row (v + 8*half), col l15.
        // NT stores: don't rinse L2 (cores stay resident), pure stream-out.
        #pragma unroll
        for (int v = 0; v < 8; ++v) {
            __builtin_nontemporal_store(
                acc[v], &outt[(16 * m + v + 8 * half) * 16 + l15]);
        }
    }
}

extern "C" void kernel_launch(void* const* d_in, const int* in_sizes, int n_in,
                              void* d_out, int out_size, void* d_ws, size_t ws_size,
                              hipStream_t stream) {
    const int*   indices = (const int*)d_in[0];
    const float* core0   = (const float*)d_in[1];
    const float* core1   = (const float*)d_in[2];
    const float* core2   = (const float*)d_in[3];
    float*       out     = (float*)d_out;

    const int T = in_sizes[0];                 // 4096*64 = 262144 tokens
    const int blocks = (T + 7) / 8;            // 8 waves (tokens) per block

    hipLaunchKernelGGL(FactorizedEmbedding_tt_kernel,
                       dim3(blocks), dim3(256), 0, stream,
                       indices, core0, core1, core2, out, T);
}